// MPNNPropPred_2259152797779
// MI455X (gfx1250) — compile-verified
//
#include <hip/hip_runtime.h>
#include <math.h>

#define N_NODES 20000
#define N_EDGES 50000
#define BGRAPHS 128
#define HDIM    64
#define FEATD   28
#define NTYPES  100
#define EDIM    5
#define MP_ITERS 3
#define S2S_ITERS 4

#define NTILES   (N_NODES / 16)   // 1250
#define WPB      8                // waves per block for GEMM kernels
#define GEMM_BLK ((NTILES + WPB - 1) / WPB)

typedef float v2f __attribute__((ext_vector_type(2)));
typedef float v8f __attribute__((ext_vector_type(8)));

__device__ __forceinline__ float sigm(float x) { return 1.0f / (1.0f + expf(-x)); }

// ---------------------------------------------------------------------------
// WMMA helpers (V_WMMA_F32_16X16X4_F32, wave32).  Fragment layouts per CDNA5
// ISA 7.12.2:
//   A 16x4 f32 : lane l holds row M=l%16; VGPR v holds K = 4*ks + 2*(l/16) + v
//   B 4x16 f32 : lane l holds col N=l%16; VGPR v holds K-row = same formula
//   C/D 16x16  : VGPR v, lane l -> row = v + 8*(l/16), col = l%16
// B is staged in LDS pair-interleaved: Bs[(k/2)*128 + col*2 + (k&1)], so a
// B fragment (two consecutive K rows, one column) is a single aligned b64.
// ---------------------------------------------------------------------------
__device__ __forceinline__ void wave_gemm_lds(const float* __restrict__ A, int lda,
                                              const float* __restrict__ Bs,
                                              int ksteps, v8f acc[4]) {
  const int lane = threadIdx.x & 31;
  const int half = lane >> 4;
  const int lr   = lane & 15;
  for (int ks = 0; ks < ksteps; ++ks) {
    const int kb = ks * 4 + half * 2;                 // even
    v2f a = *(const v2f*)(A + lr * lda + kb);         // global b64
#pragma unroll
    for (int n = 0; n < 4; ++n) {
      v2f b = *(const v2f*)(Bs + (kb >> 1) * 128 + (n * 16 + lr) * 2);  // ds b64
      acc[n] = __builtin_amdgcn_wmma_f32_16x16x4_f32(
          false, a, false, b, (short)0, acc[n], false, false);
    }
  }
}

// stage a [krows x 64] row-major global matrix into pair-interleaved LDS
__device__ __forceinline__ void stage_B(const float* __restrict__ Bglob,
                                        float* __restrict__ Bs, int krows) {
  const int total = krows * HDIM;                     // multiple of 256
  for (int base = threadIdx.x * 4; base < total; base += blockDim.x * 4) {
    const float4 v = *(const float4*)(Bglob + base); // coalesced b128
    const int k = base >> 6, col = base & 63;
    float* dst = &Bs[(k >> 1) * 128 + (k & 1)];
    dst[(col + 0) * 2] = v.x;
    dst[(col + 1) * 2] = v.y;
    dst[(col + 2) * 2] = v.z;
    dst[(col + 3) * 2] = v.w;
  }
}

// ---- edge-type weight tables: Wt[t] = relu(W_e1[t]+b_e1) @ W_e2 + b_e2 -----
__global__ void k_edge_weights(const float* __restrict__ W_e1, const float* __restrict__ b_e1,
                               const float* __restrict__ W_e2, const float* __restrict__ b_e2,
                               float* __restrict__ Wt) {
  __shared__ float hid[HDIM];
  const int t = blockIdx.x;       // 0..4
  const int j = threadIdx.x;      // 0..63
  hid[j] = fmaxf(W_e1[t * HDIM + j] + b_e1[j], 0.0f);
  __syncthreads();
  for (int q = 0; q < HDIM; ++q) {
    const int m = q * HDIM + j;   // coalesced over j
    float acc = b_e2[m];
    for (int k = 0; k < HDIM; ++k) acc += hid[k] * W_e2[k * (HDIM * HDIM) + m];
    Wt[t * (HDIM * HDIM) + m] = acc;
  }
}

// ---- node embedding: h = one_hot(type)@W_emb[:100] + feat@W_emb[100:] + b --
__global__ void k_embed(const float* __restrict__ feat, const int* __restrict__ ntype,
                        const float* __restrict__ W_emb, const float* __restrict__ b_emb,
                        float* __restrict__ h) {
  __shared__ float Bs[(FEATD / 2) * 128];             // 14*128 floats = 7 KB
  const int wid  = threadIdx.x >> 5;
  const int tile = blockIdx.x * WPB + wid;
  const bool act = tile < NTILES;
  const float* A = feat + (size_t)(act ? tile : 0) * 16 * FEATD;
  if (act) __builtin_prefetch(A, 0, 0);
  stage_B(W_emb + NTYPES * HDIM, Bs, FEATD);
  __syncthreads();
  if (!act) return;
  v8f acc[4] = {};
  wave_gemm_lds(A, FEATD, Bs, FEATD / 4, acc);
  const int lane = threadIdx.x & 31;
  const int half = lane >> 4, lr = lane & 15;
#pragma unroll
  for (int n = 0; n < 4; ++n)
#pragma unroll
    for (int v = 0; v < 8; ++v) {
      const int row = tile * 16 + v + 8 * half;
      const int col = n * 16 + lr;
      h[(size_t)row * HDIM + col] =
          acc[n][v] + W_emb[(size_t)ntype[row] * HDIM + col] + b_emb[col];
    }
}

// ---- M_t = h @ Wt[t] for all 5 types; A tile stays hot across types --------
__global__ void k_msg(const float* __restrict__ h, const float* __restrict__ Wt,
                      float* __restrict__ M) {
  __shared__ float Bs[(HDIM / 2) * 128];              // 16 KB
  const int wid  = threadIdx.x >> 5;
  const int tile = blockIdx.x * WPB + wid;
  const bool act = tile < NTILES;
  const float* A = h + (size_t)(act ? tile : 0) * 16 * HDIM;
  if (act) __builtin_prefetch(A, 0, 0);
  const int lane = threadIdx.x & 31;
  const int half = lane >> 4, lr = lane & 15;
  for (int t = 0; t < EDIM; ++t) {
    __syncthreads();
    stage_B(Wt + t * (HDIM * HDIM), Bs, HDIM);
    __syncthreads();
    if (act) {
      v8f acc[4] = {};
      wave_gemm_lds(A, HDIM, Bs, HDIM / 4, acc);
      float* out = M + ((size_t)t * N_NODES + (size_t)tile * 16) * HDIM;
#pragma unroll
      for (int n = 0; n < 4; ++n)
#pragma unroll
        for (int v = 0; v < 8; ++v)
          out[(v + 8 * half) * HDIM + n * 16 + lr] = acc[n][v];
    }
  }
}

// ---- h_new = h @ roots[i] + conv_bias[i] -----------------------------------
__global__ void k_root(const float* __restrict__ h, const float* __restrict__ root,
                       const float* __restrict__ cbias, float* __restrict__ hn) {
  __shared__ float Bs[(HDIM / 2) * 128];              // 16 KB
  const int wid  = threadIdx.x >> 5;
  const int tile = blockIdx.x * WPB + wid;
  const bool act = tile < NTILES;
  const float* A = h + (size_t)(act ? tile : 0) * 16 * HDIM;
  if (act) __builtin_prefetch(A, 0, 0);
  stage_B(root, Bs, HDIM);
  __syncthreads();
  if (!act) return;
  v8f acc[4] = {};
  wave_gemm_lds(A, HDIM, Bs, HDIM / 4, acc);
  float* out = hn + (size_t)tile * 16 * HDIM;
  const int lane = threadIdx.x & 31;
  const int half = lane >> 4, lr = lane & 15;
#pragma unroll
  for (int n = 0; n < 4; ++n)
#pragma unroll
    for (int v = 0; v < 8; ++v) {
      const int col = n * 16 + lr;
      out[(v + 8 * half) * HDIM + col] = acc[n][v] + cbias[col];
    }
}

// ---- scatter-add:  h_new[dst] += M[type][src] ------------------------------
__global__ void k_scatter(const int* __restrict__ eidx, const int* __restrict__ etype,
                          const float* __restrict__ M, float* __restrict__ hn) {
  const int tid = blockIdx.x * blockDim.x + threadIdx.x;
  const int e = tid >> 6, j = tid & 63;
  if (e >= N_EDGES) return;
  const int s = eidx[e], d = eidx[N_EDGES + e], t = etype[e];
  const float val = M[((size_t)t * N_NODES + s) * HDIM + j];
  atomicAdd(&hn[(size_t)d * HDIM + j], val);
}

// ---- Set2Set state init + per-graph ranges ---------------------------------
__global__ void k_init(float* hx, float* cx, float* qs, int* counts) {
  const int i = blockIdx.x * 256 + threadIdx.x;
  if (i < BGRAPHS * HDIM) { hx[i] = 0.0f; cx[i] = 0.0f; }
  if (i < BGRAPHS * 2 * HDIM) qs[i] = 0.0f;
  if (i < BGRAPHS) counts[i] = 0;
}

__global__ void k_count(const int* __restrict__ batch, int* counts) {
  const int i = blockIdx.x * 256 + threadIdx.x;
  if (i < N_NODES) atomicAdd(&counts[batch[i]], 1);
}

__global__ void k_scan(const int* __restrict__ counts, int* offs) {
  if (threadIdx.x == 0) {
    int a = 0;
    for (int b = 0; b < BGRAPHS; ++b) { offs[b] = a; a += counts[b]; }
    offs[BGRAPHS] = a;
  }
}

// ---- LSTM cell step (block per graph, 256 threads = 4H gates) --------------
__global__ void k_lstm(const float* __restrict__ qs, float* hx, float* cx,
                       const float* __restrict__ W_ih, const float* __restrict__ W_hh,
                       const float* __restrict__ b_ih, const float* __restrict__ b_hh) {
  __shared__ float g[4 * HDIM];
  const int b = blockIdx.x, j = threadIdx.x;  // j in [0,256)
  const float* q = qs + (size_t)b * 2 * HDIM;
  const float* hh = hx + (size_t)b * HDIM;
  float acc = b_ih[j] + b_hh[j];
  for (int k = 0; k < 2 * HDIM; ++k) acc += q[k] * W_ih[(size_t)j * 2 * HDIM + k];
  for (int k = 0; k < HDIM; ++k)     acc += hh[k] * W_hh[(size_t)j * HDIM + k];
  g[j] = acc;
  __syncthreads();
  if (j < HDIM) {
    const float ig = g[j], fg = g[HDIM + j], gg = g[2 * HDIM + j], og = g[3 * HDIM + j];
    const float c = sigm(fg) * cx[b * HDIM + j] + sigm(ig) * tanhf(gg);
    cx[b * HDIM + j] = c;
    hx[b * HDIM + j] = sigm(og) * tanhf(c);
  }
}

// ---- attention logits: e[i] = <h[i], q[batch[i]]> --------------------------
__global__ void k_dot(const float* __restrict__ h, const float* __restrict__ hx,
                      const int* __restrict__ batch, float* __restrict__ eArr) {
  const int i = blockIdx.x * 256 + threadIdx.x;
  if (i >= N_NODES) return;
  const float* hr = h + (size_t)i * HDIM;
  const float* q = hx + (size_t)batch[i] * HDIM;
  float s = 0.0f;
  for (int j = 0; j < HDIM; ++j) s += hr[j] * q[j];
  eArr[i] = s;
}

// ---- per-graph softmax pooling; writes q_star[b] = [hx[b], r[b]] -----------
__global__ void k_attn(const float* __restrict__ h, const float* __restrict__ eArr,
                       const float* __restrict__ hx, const int* __restrict__ offs,
                       float* __restrict__ qs) {
  __shared__ float red[256];
  __shared__ float m_sh, d_sh;
  const int b = blockIdx.x, tid = threadIdx.x;
  const int s = offs[b], e = offs[b + 1];
  // max
  float lm = -INFINITY;
  for (int i = s + tid; i < e; i += 256) lm = fmaxf(lm, eArr[i]);
  red[tid] = lm; __syncthreads();
  for (int st = 128; st > 0; st >>= 1) {
    if (tid < st) red[tid] = fmaxf(red[tid], red[tid + st]);
    __syncthreads();
  }
  if (tid == 0) m_sh = red[0];
  __syncthreads();
  const float m = m_sh;
  // denom
  float ls = 0.0f;
  for (int i = s + tid; i < e; i += 256) ls += expf(eArr[i] - m);
  __syncthreads();
  red[tid] = ls; __syncthreads();
  for (int st = 128; st > 0; st >>= 1) {
    if (tid < st) red[tid] += red[tid + st];
    __syncthreads();
  }
  if (tid == 0) { const float d = red[0]; d_sh = (d == 0.0f) ? 1.0f : d; }
  __syncthreads();
  const float denom = d_sh;
  // weighted sum: 4 chunks x 64 columns
  const int j = tid & 63, chunk = tid >> 6;
  float rj = 0.0f;
  for (int i = s + chunk; i < e; i += 4)
    rj += (expf(eArr[i] - m) / denom) * h[(size_t)i * HDIM + j];
  __syncthreads();
  red[tid] = rj; __syncthreads();
  if (tid < HDIM) {
    const float r = red[tid] + red[64 + tid] + red[128 + tid] + red[192 + tid];
    qs[(size_t)b * 2 * HDIM + HDIM + tid] = r;
    qs[(size_t)b * 2 * HDIM + tid] = hx[b * HDIM + tid];
  }
}

// ---- output MLP: relu(q_star @ W_o1 + b_o1) @ W_o2 + b_o2 ------------------
__global__ void k_out(const float* __restrict__ qs, const float* __restrict__ W_o1,
                      const float* __restrict__ b_o1, const float* __restrict__ W_o2,
                      const float* __restrict__ b_o2, float* __restrict__ out) {
  __shared__ float red[HDIM];
  const int b = blockIdx.x, j = threadIdx.x;  // 64 threads
  const float* q = qs + (size_t)b * 2 * HDIM;
  float acc = b_o1[j];
  for (int k = 0; k < 2 * HDIM; ++k) acc += q[k] * W_o1[k * HDIM + j];
  acc = fmaxf(acc, 0.0f);
  red[j] = acc * W_o2[j];
  __syncthreads();
  for (int st = 32; st > 0; st >>= 1) {
    if (j < st) red[j] += red[j + st];
    __syncthreads();
  }
  if (j == 0) out[b] = red[0] + b_o2[0];
}

extern "C" void kernel_launch(void* const* d_in, const int* in_sizes, int n_in,
                              void* d_out, int out_size, void* d_ws, size_t ws_size,
                              hipStream_t stream) {
  (void)in_sizes; (void)n_in; (void)out_size; (void)ws_size;
  const float* node_feat  = (const float*)d_in[0];
  const int*   node_type  = (const int*)d_in[1];
  const int*   edge_index = (const int*)d_in[2];
  const int*   edge_type  = (const int*)d_in[3];
  const int*   batch      = (const int*)d_in[4];
  const float* W_emb = (const float*)d_in[5];
  const float* b_emb = (const float*)d_in[6];
  const float* W_e1  = (const float*)d_in[7];
  const float* b_e1  = (const float*)d_in[8];
  const float* W_e2  = (const float*)d_in[9];
  const float* b_e2  = (const float*)d_in[10];
  const float* roots = (const float*)d_in[11];
  const float* cbias = (const float*)d_in[12];
  const float* W_ih  = (const float*)d_in[13];
  const float* W_hh  = (const float*)d_in[14];
  const float* b_ih  = (const float*)d_in[15];
  const float* b_hh  = (const float*)d_in[16];
  const float* W_o1  = (const float*)d_in[17];
  const float* b_o1  = (const float*)d_in[18];
  const float* W_o2  = (const float*)d_in[19];
  const float* b_o2  = (const float*)d_in[20];

  // scratch layout (floats)
  float* w    = (float*)d_ws;
  float* Wt   = w;                                       // 5*64*64
  float* hA   = Wt + EDIM * HDIM * HDIM;                 // N*64
  float* hB   = hA + (size_t)N_NODES * HDIM;             // N*64
  float* M    = hB + (size_t)N_NODES * HDIM;             // 5*N*64
  float* eArr = M + (size_t)EDIM * N_NODES * HDIM;       // N
  float* hx   = eArr + N_NODES;                          // B*64
  float* cx   = hx + BGRAPHS * HDIM;                     // B*64
  float* qs   = cx + BGRAPHS * HDIM;                     // B*128
  int*   counts = (int*)(qs + BGRAPHS * 2 * HDIM);       // B
  int*   offs   = counts + BGRAPHS;                      // B+1

  k_edge_weights<<<EDIM, HDIM, 0, stream>>>(W_e1, b_e1, W_e2, b_e2, Wt);
  k_embed<<<GEMM_BLK, WPB * 32, 0, stream>>>(node_feat, node_type, W_emb, b_emb, hA);

  float* hcur = hA; float* hnext = hB;
  for (int it = 0; it < MP_ITERS; ++it) {
    k_msg<<<GEMM_BLK, WPB * 32, 0, stream>>>(hcur, Wt, M);
    k_root<<<GEMM_BLK, WPB * 32, 0, stream>>>(hcur, roots + it * HDIM * HDIM,
                                              cbias + it * HDIM, hnext);
    k_scatter<<<(N_EDGES * HDIM + 255) / 256, 256, 0, stream>>>(edge_index, edge_type,
                                                                M, hnext);
    float* t = hcur; hcur = hnext; hnext = t;
  }

  k_init<<<(BGRAPHS * 2 * HDIM + 255) / 256, 256, 0, stream>>>(hx, cx, qs, counts);
  k_count<<<(N_NODES + 255) / 256, 256, 0, stream>>>(batch, counts);
  k_scan<<<1, 32, 0, stream>>>(counts, offs);

  for (int s2 = 0; s2 < S2S_ITERS; ++s2) {
    k_lstm<<<BGRAPHS, 256, 0, stream>>>(qs, hx, cx, W_ih, W_hh, b_ih, b_hh);
    k_dot<<<(N_NODES + 255) / 256, 256, 0, stream>>>(hcur, hx, batch, eArr);
    k_attn<<<BGRAPHS, 256, 0, stream>>>(hcur, eArr, hx, offs, qs);
  }

  k_out<<<BGRAPHS, HDIM, 0, stream>>>(qs, W_o1, b_o1, W_o2, b_o2, (float*)d_out);
}